// Memory_fingerprint_64776696758288
// MI455X (gfx1250) — compile-verified
//
#include <hip/hip_runtime.h>
#include <math.h>

typedef float v2f __attribute__((ext_vector_type(2)));
typedef float v4f __attribute__((ext_vector_type(4)));
typedef float v8f __attribute__((ext_vector_type(8)));

#define BB 64
#define LL 64
#define DD 512
#define MM 6304            // memory-context rows (== 403456/64), 6304 = 394*16
#define KTOP 50
#define EPSN 1e-8f

// ---------------------------------------------------------------------------
// Kernel A: per-row inverse clamped L2 norms for q (64 rows) and mc (6304 rows)
// grid = 64 + 6304 blocks, 128 threads (each thread one float4 of the row)
// ---------------------------------------------------------------------------
__global__ void rownorm_kernel(const float* __restrict__ q,
                               const float* __restrict__ mc,
                               float* __restrict__ qinv,
                               float* __restrict__ minv) {
  int row = blockIdx.x;
  const float* src;
  float* dst;
  if (row < BB) { src = q  + (size_t)row * DD;        dst = qinv + row; }
  else          { src = mc + (size_t)(row - BB) * DD; dst = minv + (row - BB); }
  int t = threadIdx.x;                       // 128 threads * 4 floats = 512
  v4f v = ((const v4f*)src)[t];
  float s = v.x * v.x + v.y * v.y + v.z * v.z + v.w * v.w;
  __shared__ float red[128];
  red[t] = s;
  __syncthreads();
  for (int stp = 64; stp > 0; stp >>= 1) {
    if (t < stp) red[t] += red[t + stp];
    __syncthreads();
  }
  if (t == 0) *dst = 1.0f / fmaxf(sqrtf(red[0]), EPSN);
}

// ---------------------------------------------------------------------------
// Kernel B: cos_sim[64,6304] = (q @ mc^T) scaled by qinv[b]*minv[m]
// One wave (32 threads) per 16x16 output tile, fp32 WMMA, K accumulated 4/step.
// A lane layout (16x4 f32): lane L holds q[btile + L%16, k0 + 2*(L/16) + {0,1}]
// B lane layout (4x16 f32): lane L holds mc[mtile + L%16, k0 + 2*(L/16) + {0,1}]
// D layout: VGPR r, lane L -> row btile + r + 8*(L/16), col mtile + L%16
// ---------------------------------------------------------------------------
__global__ void cosgemm_kernel(const float* __restrict__ q,
                               const float* __restrict__ mc,
                               const float* __restrict__ qinv,
                               const float* __restrict__ minv,
                               float* __restrict__ cosm) {
  int mtile = blockIdx.x * 16;               // 394 tiles
  int btile = blockIdx.y * 16;               // 4 tiles
  int lane   = threadIdx.x;                  // 32 threads = 1 wave
  int laneLo = lane & 15;
  int laneHi = lane >> 4;

  const float* qrow = q  + (size_t)(btile + laneLo) * DD + 2 * laneHi;
  const float* mrow = mc + (size_t)(mtile + laneLo) * DD + 2 * laneHi;

  v8f c = {};
#pragma unroll 4
  for (int k0 = 0; k0 < DD; k0 += 4) {
    v2f a = *(const v2f*)(qrow + k0);
    v2f b = *(const v2f*)(mrow + k0);
    // D = A(16x4 f32) x B(4x16 f32) + C
    c = __builtin_amdgcn_wmma_f32_16x16x4_f32(
        /*neg_a=*/false, a, /*neg_b=*/false, b,
        /*c_mod=*/(short)0, c, /*reuse_a=*/false, /*reuse_b=*/false);
  }

  int col = mtile + laneLo;
  float mi = minv[col];
#pragma unroll
  for (int r = 0; r < 8; ++r) {
    int row = btile + r + 8 * laneHi;
    cosm[(size_t)row * MM + col] = c[r] * qinv[row] * mi;
  }
}

// ---------------------------------------------------------------------------
// Kernel C: per-batch top-50 (iterative argmax in LDS, lowest-index tiebreak
// to match jax.lax.top_k) + softmax over the (descending-sorted) values.
// grid = 64 blocks, 256 threads. LDS: 6304 floats row copy (~25 KB).
// ---------------------------------------------------------------------------
__global__ void topk_softmax_kernel(const float* __restrict__ cosm,
                                    float* __restrict__ w_out,
                                    int* __restrict__ idx_out) {
  int b = blockIdx.x;
  int t = threadIdx.x;                       // 256
  __shared__ float sv[MM];
  __shared__ float rv[256];
  __shared__ int   ri[256];
  __shared__ float tkv[KTOP];
  __shared__ int   tki[KTOP];
  __shared__ float ssum;

  const float* row = cosm + (size_t)b * MM;
  for (int i = t; i < MM; i += 256) sv[i] = row[i];
  __syncthreads();

  for (int iter = 0; iter < KTOP; ++iter) {
    float bv = -INFINITY;
    int   bi = MM;
    for (int i = t; i < MM; i += 256) {
      float v = sv[i];
      if (v > bv) { bv = v; bi = i; }        // strictly '>' keeps lowest index
    }
    rv[t] = bv; ri[t] = bi;
    __syncthreads();
    for (int s = 128; s > 0; s >>= 1) {
      if (t < s) {
        float ov = rv[t + s]; int oi = ri[t + s];
        if (ov > rv[t] || (ov == rv[t] && oi < ri[t])) { rv[t] = ov; ri[t] = oi; }
      }
      __syncthreads();
    }
    if (t == 0) {
      tkv[iter] = rv[0];
      tki[iter] = ri[0];
      sv[ri[0]] = -INFINITY;                 // remove the winner
    }
    __syncthreads();
  }

  // softmax: values are sorted descending, so tkv[0] is the max
  if (t < KTOP) rv[t] = expf(tkv[t] - tkv[0]);
  __syncthreads();
  if (t == 0) {
    float s = 0.0f;
    for (int i = 0; i < KTOP; ++i) s += rv[i];
    ssum = s;
  }
  __syncthreads();
  if (t < KTOP) {
    w_out[b * KTOP + t]   = rv[t] / ssum;
    idx_out[b * KTOP + t] = tki[t];
  }
}

// ---------------------------------------------------------------------------
// Kernel D: out[b,l,:] = weight * sum_k w[b,k]*fp[idx[b,k]*64 + l, :]
//                      + (1-weight) * enc[b,l,:]
// grid = B*L = 4096 blocks, 128 threads (one float4 of D each). ~419 MB of
// HBM reads -> this is the bandwidth-bound hot kernel (~18 us at 23.3 TB/s).
// ---------------------------------------------------------------------------
__global__ void gather_blend_kernel(const float* __restrict__ fp,
                                    const float* __restrict__ enc,
                                    const float* __restrict__ w_in,
                                    const int* __restrict__ idx_in,
                                    const float* __restrict__ wscalar,
                                    float* __restrict__ out) {
  int bl = blockIdx.x;                       // 0..4095
  int b  = bl >> 6;
  int l  = bl & 63;
  int t  = threadIdx.x;                      // 128 -> float4 lanes over D=512

  __shared__ float ws[KTOP];
  __shared__ int   is[KTOP];
  if (t < KTOP) {
    ws[t] = w_in[b * KTOP + t];
    is[t] = idx_in[b * KTOP + t];
  }
  __syncthreads();

  v4f acc = {};
#pragma unroll 2
  for (int k = 0; k < KTOP; ++k) {
    size_t rowoff = ((size_t)is[k] * LL + l) * (DD / 4);
    v4f v = ((const v4f*)fp)[rowoff + t];
    float w = ws[k];
    acc.x += w * v.x; acc.y += w * v.y; acc.z += w * v.z; acc.w += w * v.w;
  }

  float wg = wscalar[0];
  float og = 1.0f - wg;
  size_t o = (size_t)bl * (DD / 4) + t;
  v4f e = ((const v4f*)enc)[o];
  v4f r;
  r.x = acc.x * wg + e.x * og;
  r.y = acc.y * wg + e.y * og;
  r.z = acc.z * wg + e.z * og;
  r.w = acc.w * wg + e.w * og;
  ((v4f*)out)[o] = r;
}

// ---------------------------------------------------------------------------
extern "C" void kernel_launch(void* const* d_in, const int* in_sizes, int n_in,
                              void* d_out, int out_size, void* d_ws, size_t ws_size,
                              hipStream_t stream) {
  const float* enc = (const float*)d_in[0];   // [64,64,512]
  const float* q   = (const float*)d_in[1];   // [64,512]
  const float* fp  = (const float*)d_in[2];   // [403456,512]
  const float* mc  = (const float*)d_in[3];   // [6304,512]
  const float* wsc = (const float*)d_in[4];   // scalar weight
  // d_in[5] = k (50), hardcoded

  // workspace layout (16B aligned)
  char* ws = (char*)d_ws;
  float* qinv  = (float*)(ws + 0);                       //    64 f
  float* minv  = (float*)(ws + 256);                     //  6304 f
  float* cosm  = (float*)(ws + 25472);                   // 64*6304 f
  float* topw  = (float*)(ws + 25472 + 1613824);         // 64*50 f
  int*   topi  = (int*)  (ws + 25472 + 1613824 + 12800); // 64*50 i
  float* out   = (float*)d_out;

  // A: norms
  rownorm_kernel<<<dim3(BB + MM), dim3(128), 0, stream>>>(q, mc, qinv, minv);
  // B: WMMA fp32 cosine-similarity GEMM
  cosgemm_kernel<<<dim3(MM / 16, BB / 16), dim3(32), 0, stream>>>(q, mc, qinv, minv, cosm);
  // C: top-50 + softmax per batch row
  topk_softmax_kernel<<<dim3(BB), dim3(256), 0, stream>>>(cosm, topw, topi);
  // D: weighted block gather + blend (bandwidth-bound)
  gather_blend_kernel<<<dim3(BB * LL), dim3(128), 0, stream>>>(fp, enc, topw, topi, wsc, out);
}